// MultiHeadSelfAttention_58454504898881
// MI455X (gfx1250) — compile-verified
//
#include <hip/hip_runtime.h>
#include <math.h>

// Problem constants (fixed by the reference)
#define BB 2
#define TT 2048
#define EE 1024
#define NH 16
#define HD 64

typedef __attribute__((ext_vector_type(2))) float v2f;
typedef __attribute__((ext_vector_type(8))) float v8f;

// ---------------------------------------------------------------------------
// Generic fp32 WMMA GEMM (V_WMMA_F32_16X16X4_F32).
//   BTF=true : C[m,n] = sum_k A[m,k] * B[n,k]   (B row-major [N x K])
//   BTF=false: C[m,n] = sum_k A[m,k] * B[k,n]   (B row-major [K x N])
// Wave tile = 64x64 (4x4 WMMA tiles, 16 v8f accumulators). Block = BWM x BWN
// waves (128 threads).
// MODE: 0 = plain store, 1 = QKV head-split store into [B,H,T,D],
//       2 = causal scores store (scale below diagonal, 0.0 above).
// TRIMK: limit K-loop to m0+64 (valid when A[m,k]==0 for k>m, e.g. causal w).
// blockIdx.z indexes (b*NH + h); per-operand offsets = (z/16)*s?b + (z%16)*s?h
// ---------------------------------------------------------------------------
template <int BWM, int BWN, bool BTF, int MODE, bool TRIMK>
__global__ __launch_bounds__(128) void gemm_wmma_kernel(
    const float* __restrict__ A, const float* __restrict__ Bm,
    float* __restrict__ C, int lda, int ldb, int ldc, int K,
    long long sAb, long long sAh, long long sBb, long long sBh,
    long long sCb, long long sCh, float scale) {
  const int lane = threadIdx.x & 31;
  const int wid = threadIdx.x >> 5;
  const int wm = wid / BWN;
  const int wn = wid % BWN;

  const int z = blockIdx.z;
  const int zb = z >> 4;  // NH == 16
  const int zh = z & 15;
  const long long offA = (long long)zb * sAb + (long long)zh * sAh;
  const long long offB = (long long)zb * sBb + (long long)zh * sBh;
  const long long offC = (long long)zb * sCb + (long long)zh * sCh;

  const int m0 = blockIdx.y * (BWM * 64) + wm * 64;
  const int n0 = blockIdx.x * (BWN * 64) + wn * 64;

  const int hh = lane >> 4;   // 0: K pair {0,1}, 1: K pair {2,3}
  const int ml = lane & 15;   // row (A) / col (B) within 16
  const int koff = hh * 2;

  v8f acc[4][4];
#pragma unroll
  for (int mi = 0; mi < 4; ++mi)
#pragma unroll
    for (int ni = 0; ni < 4; ++ni) acc[mi][ni] = {};

  // Causal-trimmed K extent: rows in this wave tile only see k <= m0+63.
  const int Keff = TRIMK ? ((K < m0 + 64) ? K : m0 + 64) : K;

  // Fully-masked causal score tile: skip the GEMM (wave-uniform branch).
  const bool skip = (MODE == 2) && (n0 > m0 + 63);
  if (!skip) {
    const float* Ap[4];
    const float* Bp[4];
#pragma unroll
    for (int mi = 0; mi < 4; ++mi)
      Ap[mi] = A + offA + (long long)(m0 + mi * 16 + ml) * lda + koff;
#pragma unroll
    for (int ni = 0; ni < 4; ++ni) {
      if (BTF)
        Bp[ni] = Bm + offB + (long long)(n0 + ni * 16 + ml) * ldb + koff;
      else
        Bp[ni] = Bm + offB + (long long)koff * ldb + (n0 + ni * 16 + ml);
    }

    for (int k0 = 0; k0 < Keff; k0 += 4) {
      v2f a[4], b[4];
#pragma unroll
      for (int mi = 0; mi < 4; ++mi) a[mi] = *(const v2f*)(Ap[mi] + k0);
#pragma unroll
      for (int ni = 0; ni < 4; ++ni) {
        if (BTF) {
          b[ni] = *(const v2f*)(Bp[ni] + k0);
        } else {
          b[ni].x = Bp[ni][(long long)k0 * ldb];
          b[ni].y = Bp[ni][(long long)k0 * ldb + ldb];
        }
      }
#pragma unroll
      for (int mi = 0; mi < 4; ++mi)
#pragma unroll
        for (int ni = 0; ni < 4; ++ni)
          acc[mi][ni] = __builtin_amdgcn_wmma_f32_16x16x4_f32(
              false, a[mi], false, b[ni], (short)0, acc[mi][ni], false, false);
    }
  }

  // Store: lane 0-15 -> N=lane, M=r; lane 16-31 -> N=lane-16, M=r+8
#pragma unroll
  for (int mi = 0; mi < 4; ++mi) {
#pragma unroll
    for (int ni = 0; ni < 4; ++ni) {
#pragma unroll
      for (int r = 0; r < 8; ++r) {
        const int m = m0 + mi * 16 + hh * 8 + r;
        const int n = n0 + ni * 16 + ml;
        float v = acc[mi][ni][r];
        if (MODE == 0) {
          C[offC + (long long)m * ldc + n] = v;
        } else if (MODE == 1) {
          // m in [0, B*T), n in [0, E): write to [B,H,T,D]
          const int bb = m >> 11, t = m & (TT - 1);
          const int h = n >> 6, d = n & (HD - 1);
          C[(((long long)bb * NH + h) * TT + t) * HD + d] = v;
        } else {
          // Masked region stores 0.0: this is already the final softmax value
          // there, so the softmax pass never needs to touch it.
          v = (n <= m) ? v * scale : 0.0f;
          C[offC + (long long)m * ldc + n] = v;
        }
      }
    }
  }
}

// ---------------------------------------------------------------------------
// Causal row softmax over T=2048 in place. One 256-thread block per row.
// Only the live prefix (ceil((q+1)/4) float4 chunks) is read/written; the
// causal mask is applied analytically (n > q -> 0). Chunks beyond the prefix
// already hold 0 from the scores kernel.
// ---------------------------------------------------------------------------
__global__ __launch_bounds__(256) void softmax_kernel(float* __restrict__ w) {
  __shared__ float sd[256];
  const long long row = blockIdx.x;
  const int q = (int)(row & (TT - 1));   // query index within this (b,h)
  const int nvec = (q >> 2) + 1;         // float4 chunks containing live data
  float4* pv = (float4*)(w + row * (long long)TT);
  const int tid = threadIdx.x;
  const int c0 = tid, c1 = tid + 256;
  const bool p0 = c0 < nvec, p1 = c1 < nvec;

  float4 v0 = p0 ? pv[c0] : make_float4(0.f, 0.f, 0.f, 0.f);
  float4 v1 = p1 ? pv[c1] : make_float4(0.f, 0.f, 0.f, 0.f);

  const float NINF = -__builtin_inff();
  const int b0 = c0 * 4, b1 = c1 * 4;
  // valid(n) = (n <= q) within a present chunk
  float m00 = (p0 && b0 + 0 <= q) ? v0.x : NINF;
  float m01 = (p0 && b0 + 1 <= q) ? v0.y : NINF;
  float m02 = (p0 && b0 + 2 <= q) ? v0.z : NINF;
  float m03 = (p0 && b0 + 3 <= q) ? v0.w : NINF;
  float m10 = (p1 && b1 + 0 <= q) ? v1.x : NINF;
  float m11 = (p1 && b1 + 1 <= q) ? v1.y : NINF;
  float m12 = (p1 && b1 + 2 <= q) ? v1.z : NINF;
  float m13 = (p1 && b1 + 3 <= q) ? v1.w : NINF;

  float lmax = fmaxf(fmaxf(fmaxf(m00, m01), fmaxf(m02, m03)),
                     fmaxf(fmaxf(m10, m11), fmaxf(m12, m13)));
  sd[tid] = lmax;
  __syncthreads();
  for (int s = 128; s > 0; s >>= 1) {
    if (tid < s) sd[tid] = fmaxf(sd[tid], sd[tid + s]);
    __syncthreads();
  }
  const float rmax = sd[0];
  __syncthreads();

  v0.x = (p0 && b0 + 0 <= q) ? expf(v0.x - rmax) : 0.0f;
  v0.y = (p0 && b0 + 1 <= q) ? expf(v0.y - rmax) : 0.0f;
  v0.z = (p0 && b0 + 2 <= q) ? expf(v0.z - rmax) : 0.0f;
  v0.w = (p0 && b0 + 3 <= q) ? expf(v0.w - rmax) : 0.0f;
  v1.x = (p1 && b1 + 0 <= q) ? expf(v1.x - rmax) : 0.0f;
  v1.y = (p1 && b1 + 1 <= q) ? expf(v1.y - rmax) : 0.0f;
  v1.z = (p1 && b1 + 2 <= q) ? expf(v1.z - rmax) : 0.0f;
  v1.w = (p1 && b1 + 3 <= q) ? expf(v1.w - rmax) : 0.0f;

  float lsum = (v0.x + v0.y) + (v0.z + v0.w) + (v1.x + v1.y) + (v1.z + v1.w);
  sd[tid] = lsum;
  __syncthreads();
  for (int s = 128; s > 0; s >>= 1) {
    if (tid < s) sd[tid] += sd[tid + s];
    __syncthreads();
  }
  const float inv = 1.0f / sd[0];

  if (p0) {
    v0.x *= inv; v0.y *= inv; v0.z *= inv; v0.w *= inv;
    pv[c0] = v0;
  }
  if (p1) {
    v1.x *= inv; v1.y *= inv; v1.z *= inv; v1.w *= inv;
    pv[c1] = v1;
  }
}

extern "C" void kernel_launch(void* const* d_in, const int* in_sizes, int n_in,
                              void* d_out, int out_size, void* d_ws,
                              size_t ws_size, hipStream_t stream) {
  (void)in_sizes; (void)n_in; (void)out_size; (void)ws_size;

  const float* x = (const float*)d_in[0];
  // d_in[1] = causal mask (recomputed analytically; unused)
  const float* WQ = (const float*)d_in[2];
  const float* WK = (const float*)d_in[3];
  const float* WV = (const float*)d_in[4];
  const float* WO = (const float*)d_in[5];

  float* y = (float*)d_out;                      // [B,T,E]
  float* w = y + (size_t)BB * TT * EE;           // [B,H,T,T]

  const size_t QKV = (size_t)BB * NH * TT * HD;  // 4,194,304 floats
  float* Qh = (float*)d_ws;                      // [B,H,T,D]
  float* Kh = Qh + QKV;
  float* Vh = Kh + QKV;
  float* merged = Qh;  // reuse Q region for merged [B,T,E] (Q dead by then)

  const dim3 blk(128);
  const long long sTD = (long long)TT * HD;      // 131072
  const long long sT2 = (long long)TT * TT;      // 4194304

  // 1) QKV projections: [4096x1024] @ [1024x1024]^T -> head-split [B,H,T,D]
  gemm_wmma_kernel<2, 2, true, 1, false><<<dim3(8, 32, 1), blk, 0, stream>>>(
      x, WQ, Qh, EE, EE, 0, EE, 0, 0, 0, 0, 0, 0, 1.0f);
  gemm_wmma_kernel<2, 2, true, 1, false><<<dim3(8, 32, 1), blk, 0, stream>>>(
      x, WK, Kh, EE, EE, 0, EE, 0, 0, 0, 0, 0, 0, 1.0f);
  gemm_wmma_kernel<2, 2, true, 1, false><<<dim3(8, 32, 1), blk, 0, stream>>>(
      x, WV, Vh, EE, EE, 0, EE, 0, 0, 0, 0, 0, 0, 1.0f);

  // 2) scores = Q K^T / sqrt(D), causal (0 above diagonal), per (b,h)
  gemm_wmma_kernel<2, 2, true, 2, false>
      <<<dim3(16, 16, BB * NH), blk, 0, stream>>>(
          Qh, Kh, w, HD, HD, TT, HD, NH * sTD, sTD, NH * sTD, sTD, NH * sT2,
          sT2, 0.125f);

  // 3) causal softmax over rows of w (prefix-only traffic)
  softmax_kernel<<<dim3(BB * NH * TT), dim3(256), 0, stream>>>(w);

  // 4) out = w @ V per (b,h), K trimmed to the causal prefix
  gemm_wmma_kernel<4, 1, false, 0, true>
      <<<dim3(1, 8, BB * NH), blk, 0, stream>>>(
          w, Vh, merged, TT, HD, EE, TT, NH * sT2, sT2, NH * sTD, sTD,
          (long long)TT * EE, HD, 1.0f);

  // 5) y = merged @ WO^T: [4096x1024] @ [1024x1024]^T
  gemm_wmma_kernel<2, 2, true, 0, false><<<dim3(8, 32, 1), blk, 0, stream>>>(
      merged, WO, y, EE, EE, EE, EE, 0, 0, 0, 0, 0, 0, 1.0f);
}